// ParcelPooling_66082366816677
// MI455X (gfx1250) — compile-verified
//
#include <hip/hip_runtime.h>

// ---------------- problem constants (match reference) ----------------
#define P_SEG   4096
#define B_DIM   4
#define C_DIM   128
#define H_DIM   512
#define W_DIM   512
#define HW_     (H_DIM * W_DIM)          // 262144 = 1<<18
#define HW_LOG2 18
#define NPIX    (B_DIM * HW_)            // 1048576 pixels

// ---------------- tuning ----------------
#define CC        16                     // channels per chunk
#define NCHUNK    (C_DIM / CC)           // 8
#define KWG       32                     // workgroups per channel chunk
#define HIST_STR  17                     // padded stride: odd -> all 64 LDS banks used
#define SCATTER_THREADS 1024
#define LDS_BYTES ((size_t)P_SEG * HIST_STR * sizeof(float))   // 278528 B = 272 KB
#define CNT_WGS   64
#define TILE_PIX  8192                   // pid pixels staged per TDM transfer (32 KB)

// ---------------- CDNA5 TDM support (guarded; plain-load fallback) ----
#if defined(__HIP_DEVICE_COMPILE__) && defined(__gfx1250__) && \
    __has_builtin(__builtin_amdgcn_tensor_load_to_lds) &&      \
    __has_builtin(__builtin_amdgcn_s_wait_tensorcnt)
#define USE_TDM 1
typedef unsigned int u32x4 __attribute__((ext_vector_type(4)));
typedef int          i32x4 __attribute__((ext_vector_type(4)));
typedef int          i32x8 __attribute__((ext_vector_type(8)));
#else
#define USE_TDM 0
#endif

// =====================================================================
// Kernel A: cnt[p] and bsum[p] from pid only.
// pid stripe is DMA'd into LDS by the Tensor Data Mover (one wave issues,
// s_wait_tensorcnt 0, then the whole block consumes from LDS).
// =====================================================================
__global__ __launch_bounds__(256) void pp_count_kernel(
    const int* __restrict__ pid,
    unsigned*  __restrict__ cnt,
    unsigned*  __restrict__ bsum)
{
    __shared__ unsigned cnt_l[P_SEG];
    __shared__ unsigned bsum_l[P_SEG];
#if USE_TDM
    __shared__ int pid_stage[TILE_PIX];
#endif
    for (int i = threadIdx.x; i < P_SEG; i += blockDim.x) {
        cnt_l[i]  = 0u;
        bsum_l[i] = 0u;
    }
    __syncthreads();

    const int stripe = NPIX / CNT_WGS;          // 16384 pixels per workgroup
    const int sbase  = blockIdx.x * stripe;

#if USE_TDM
    const unsigned lds_off = (unsigned)(unsigned long long)(void*)pid_stage;
    for (int t = 0; t < stripe / TILE_PIX; ++t) {      // 2 tiles of 8192
        const int tile_base = sbase + t * TILE_PIX;
        if (threadIdx.x < 32) {
            unsigned long long gaddr =
                (unsigned long long)(const void*)(pid + tile_base);
            // D# group0: count=1 | lds_addr | global_addr[56:0] | type=2
            u32x4 g0 = { 1u,
                         lds_off,
                         (unsigned)(gaddr & 0xffffffffull),
                         (unsigned)((gaddr >> 32) & 0x01ffffffull) | (2u << 30) };
            // D# group1: data_size=4B; tensor_dim0=8192 (bits 79:48);
            // tensor_dim1=1 (bits 111:80); tile_dim0=8192 (bits 127:112);
            // tile_dim1=1 (bits 143:128); tensor_dim0_stride=8192 (bits 207:160)
            i32x8 g1 = { (int)0x00020000,
                         (int)(TILE_PIX << 16),
                         (int)(1 << 16),
                         (int)(TILE_PIX << 16),
                         1,
                         TILE_PIX,
                         0, 0 };
            i32x4 gz = { 0, 0, 0, 0 };
#if __clang_major__ >= 23
            i32x8 gz8 = { 0, 0, 0, 0, 0, 0, 0, 0 };
            __builtin_amdgcn_tensor_load_to_lds(g0, g1, gz, gz, gz8, 0);
#else
            __builtin_amdgcn_tensor_load_to_lds(g0, g1, gz, gz, 0);
#endif
            __builtin_amdgcn_s_wait_tensorcnt(0);
        }
        __syncthreads();
        for (int i = threadIdx.x; i < TILE_PIX; i += blockDim.x) {
            int p = pid_stage[i];
            if (p > 0) {
                unsigned b = (unsigned)(tile_base + i) >> HW_LOG2;
                atomicAdd(&cnt_l[p], 1u);       // ds_add_u32
                atomicAdd(&bsum_l[p], b);       // ds_add_u32
            }
        }
        __syncthreads();
    }
#else
    for (int pix = sbase + (int)threadIdx.x; pix < sbase + stripe; pix += blockDim.x) {
        int p = pid[pix];
        if (p > 0) {
            unsigned b = (unsigned)pix >> HW_LOG2;
            atomicAdd(&cnt_l[p], 1u);
            atomicAdd(&bsum_l[p], b);
        }
    }
    __syncthreads();
#endif

    for (int i = threadIdx.x; i < P_SEG; i += blockDim.x) {
        unsigned c = cnt_l[i];
        if (c) {
            atomicAdd(&cnt[i],  c);
            atomicAdd(&bsum[i], bsum_l[i]);
        }
    }
}

// =====================================================================
// Kernel B: main scatter. blockIdx = chunk*KWG + sub. Each thread owns
// 4 consecutive pixels -> int4 pid load + b128 feat load per channel.
// LDS histogram (272 KB, stride 17) accumulated with ds_add_f32, then
// fp32 global-atomic merge.
// =====================================================================
__global__ __launch_bounds__(SCATTER_THREADS, 1) void pp_scatter_kernel(
    const float* __restrict__ feat,
    const int*   __restrict__ pid,
    float*       __restrict__ vsum)   // [P_SEG][C_DIM]
{
    extern __shared__ float hist[];   // [P_SEG][HIST_STR]

    const int chunk = blockIdx.x / KWG;   // 0..NCHUNK-1
    const int sub   = blockIdx.x % KWG;   // 0..KWG-1
    const int c0    = chunk * CC;

    for (int i = threadIdx.x; i < P_SEG * HIST_STR; i += blockDim.x)
        hist[i] = 0.0f;
    __syncthreads();

    const int pix_per_wg = NPIX / KWG;    // 32768
    const int start = sub * pix_per_wg;
    const int end   = start + pix_per_wg;
    const int step  = blockDim.x * 4;

    for (int base = start + (int)threadIdx.x * 4; base < end; base += step) {
        const int b   = base >> HW_LOG2;           // 4 pixels share b and plane
        const int off = base & (HW_ - 1);
        const float* fbase = feat + ((size_t)b * C_DIM + c0) * (size_t)HW_ + off;

        // prefetch next iteration's streams (global_prefetch_b8)
        int nxt = base + step;
        if (nxt < end) {
            __builtin_prefetch(pid + nxt, 0, 0);
            __builtin_prefetch(fbase + step, 0, 0);
        }

        const int4 p4 = *(const int4*)(pid + base);   // global_load_b128
        #pragma unroll
        for (int c = 0; c < CC; ++c) {
            const float4 f4 = *(const float4*)(fbase + (size_t)c * HW_); // b128
            if (p4.x > 0) atomicAdd(&hist[p4.x * HIST_STR + c], f4.x);   // ds_add_f32
            if (p4.y > 0) atomicAdd(&hist[p4.y * HIST_STR + c], f4.y);
            if (p4.z > 0) atomicAdd(&hist[p4.z * HIST_STR + c], f4.z);
            if (p4.w > 0) atomicAdd(&hist[p4.w * HIST_STR + c], f4.w);
        }
    }
    __syncthreads();

    // Merge LDS partial histogram into global accumulator (coalesced sweep).
    for (int i = threadIdx.x; i < P_SEG * CC; i += blockDim.x) {
        const int p = i / CC;
        const int c = i % CC;
        float v = hist[p * HIST_STR + c];
        if (v != 0.0f)
            atomicAdd(&vsum[(size_t)p * C_DIM + c0 + c], v);   // global_atomic_add_f32
    }
}

// =====================================================================
// Kernel C: finalize -> d_out = [ v (4095*128) | batch (4095) | pid_o (4095) ]
// =====================================================================
__global__ __launch_bounds__(256) void pp_finalize_kernel(
    const float*    __restrict__ vsum,
    const unsigned* __restrict__ cnt,
    const unsigned* __restrict__ bsum,
    float*          __restrict__ out,
    int total)
{
    const int NV = (P_SEG - 1) * C_DIM;       // 524160
    const int NB = P_SEG - 1;                 // 4095
    int i = blockIdx.x * blockDim.x + threadIdx.x;
    if (i >= total) return;

    if (i < NV) {
        int p = i / C_DIM + 1;
        int c = i % C_DIM;
        float cf = (float)cnt[p];
        out[i] = vsum[(size_t)p * C_DIM + c] / (cf + 1e-6f);
    } else if (i < NV + NB) {
        int k = i - NV;                       // 0..4094 -> pid k+1
        out[i] = (float)bsum[k + 1];
    } else {
        int k = i - NV - NB;
        out[i] = (float)(k + 1);              // pid_o = 1..4095
    }
}

// =====================================================================
extern "C" void kernel_launch(void* const* d_in, const int* in_sizes, int n_in,
                              void* d_out, int out_size, void* d_ws, size_t ws_size,
                              hipStream_t stream)
{
    (void)in_sizes; (void)n_in; (void)out_size; (void)ws_size;

    const float* feat = (const float*)d_in[0];
    const int*   pid  = (const int*)d_in[1];
    float*       out  = (float*)d_out;

    // workspace layout: vsum[P*C] f32 | cnt[P] u32 | bsum[P] u32
    float*    vsum = (float*)d_ws;
    unsigned* cnt  = (unsigned*)(vsum + (size_t)P_SEG * C_DIM);
    unsigned* bsum = cnt + P_SEG;
    const size_t acc_bytes = ((size_t)P_SEG * C_DIM + 2 * P_SEG) * sizeof(float);

    hipMemsetAsync(d_ws, 0, acc_bytes, stream);

    pp_count_kernel<<<CNT_WGS, 256, 0, stream>>>(pid, cnt, bsum);

    pp_scatter_kernel<<<NCHUNK * KWG, SCATTER_THREADS, LDS_BYTES, stream>>>(feat, pid, vsum);

    const int total = (P_SEG - 1) * C_DIM + 2 * (P_SEG - 1);
    pp_finalize_kernel<<<(total + 255) / 256, 256, 0, stream>>>(vsum, cnt, bsum, out, total);
}